// GraphAttention_5557687681686
// MI455X (gfx1250) — compile-verified
//
#include <hip/hip_runtime.h>

typedef float v2f __attribute__((ext_vector_type(2)));
typedef float v8f __attribute__((ext_vector_type(8)));

typedef __attribute__((address_space(1))) int gint_t;   // global int
typedef __attribute__((address_space(3))) int lint_t;   // LDS int

#define GAT_B   2
#define GAT_N   16384
#define GAT_U   128
#define GAT_BIN 32
#define GAT_ROWS (GAT_B * GAT_N)   // 32768 rows of h

// ---------------------------------------------------------------------------
// Kernel 1: h = node_states @ kernel  (f32 WMMA 16x16x4), fused with
//           e_src = h @ a_src, e_dst = h @ a_dst  (cross-lane reduction).
// One wave computes a 16-row x 128-col strip of h. 4 waves / block.
//
// Weight matrix staged in LDS in K-pair-interleaved layout:
//   ldsB[(k>>1)*256 + n*2 + (k&1)] = kernel[k][n]
// so each WMMA B-fragment (B[k][n], B[k+1][n]) is ONE aligned ds_load_b64
// straight into the operand register pair (no v_mov fixups).
// Staging uses the CDNA5 async global->LDS path (ASYNCcnt) when available.
// ---------------------------------------------------------------------------
__global__ __launch_bounds__(128) void gat_gemm_wmma(
    const float* __restrict__ ns,    // (B*N, U) node_states
    const float* __restrict__ kmat,  // (U, U) kernel
    const float* __restrict__ katt,  // (2U,) attention vector
    float* __restrict__ hbuf,        // (B*N, U) out
    float* __restrict__ esrc,        // (B*N,) out
    float* __restrict__ edst)        // (B*N,) out
{
    __shared__ float ldsB[GAT_U * GAT_U];   // 64 KB of the 320 KB per WGP

#if __has_builtin(__builtin_amdgcn_global_load_async_to_lds_b32)
    // Async scatter: each element goes straight to its paired LDS slot.
    for (int i = threadIdx.x; i < GAT_U * GAT_U; i += 128) {
        const int k = i >> 7;          // row of kernel
        const int n = i & 127;         // col of kernel
        const int di = ((k >> 1) * 256) + (n * 2) + (k & 1);
        __builtin_amdgcn_global_load_async_to_lds_b32(
            (gint_t*)(kmat + i),
            (lint_t*)(ldsB + di),
            0, 0);
    }
#if __has_builtin(__builtin_amdgcn_s_wait_asynccnt)
    __builtin_amdgcn_s_wait_asynccnt(0);
#else
    asm volatile("s_wait_asynccnt 0x0" ::: "memory");
#endif
#else
    for (int i = threadIdx.x; i < GAT_U * GAT_U; i += 128) {
        const int k = i >> 7;
        const int n = i & 127;
        ldsB[((k >> 1) * 256) + (n * 2) + (k & 1)] = kmat[i];
    }
#endif
    __syncthreads();

    const int lane  = threadIdx.x & 31;
    const int wv    = threadIdx.x >> 5;
    const int strip = blockIdx.x * 4 + wv;   // 16-row strip index
    const int row0  = strip * 16;
    const int m     = lane & 15;             // row (A) / col (B,C,D) within tile
    const int hi    = lane >> 4;             // half-wave select
    const int kb    = hi * 2;                // K sub-offset for A/B fragments

    v8f acc[8];
#pragma unroll
    for (int nt = 0; nt < 8; ++nt) acc[nt] = (v8f)0.f;

    // A fragment source: A[m][k+kb], A[m][k+kb+1] per K-step (8B aligned b64)
    const float* arow = ns + (size_t)(row0 + m) * GAT_U + kb;

    for (int k = 0; k < GAT_U; k += 4) {
        v2f a = *(const v2f*)(arow + k);
        // paired layout: pair index p = (k+kb)/2 (k%4==0, kb in {0,2} -> even)
        const v2f* bp = (const v2f*)(ldsB + ((k + kb) >> 1) * 256 + m * 2);
#pragma unroll
        for (int nt = 0; nt < 8; ++nt) {
            v2f bf = bp[nt * 16];   // one ds_load_b64: {B[k+kb][n], B[k+kb+1][n]}
            acc[nt] = __builtin_amdgcn_wmma_f32_16x16x4_f32(
                false, a, false, bf, (short)0, acc[nt], false, false);
        }
    }

    // Per-lane slice of attention vectors (col = nt*16 + m)
    float asv[8], adv[8];
#pragma unroll
    for (int nt = 0; nt < 8; ++nt) {
        asv[nt] = katt[nt * 16 + m];
        adv[nt] = katt[GAT_U + nt * 16 + m];
    }

    // Epilogue: store h strip + reduce e_src/e_dst per row.
    // C/D layout: acc[nt][r] = h[row0 + r + 8*hi][nt*16 + m]
#pragma unroll
    for (int r = 0; r < 8; ++r) {
        const int row = row0 + r + 8 * hi;
        float ps = 0.f, pd = 0.f;
#pragma unroll
        for (int nt = 0; nt < 8; ++nt) {
            float hv = acc[nt][r];
            hbuf[(size_t)row * GAT_U + nt * 16 + m] = hv;
            ps += hv * asv[nt];
            pd += hv * adv[nt];
        }
        // reduce across the 16 lanes of this half-wave (masks <16 stay in half)
#pragma unroll
        for (int msk = 8; msk >= 1; msk >>= 1) {
            ps += __shfl_xor(ps, msk);
            pd += __shfl_xor(pd, msk);
        }
        if (m == 0) { esrc[row] = ps; edst[row] = pd; }
    }
}

// ---------------------------------------------------------------------------
// Kernel 2: per (b, n): softmax over the node's 32 edges, then
//           out[b,n,:] = sum_j w_j * h[b, src_j, :].
// One wave per output node; lane j owns edge j; each lane accumulates a
// float4 slice (cols lane*4 .. lane*4+3) of the 128-wide output.
// h gathers are random but L2-resident (h = 16 MB << 192 MB L2).
// ---------------------------------------------------------------------------
__global__ __launch_bounds__(256) void gat_aggregate(
    const float* __restrict__ h,     // (B*N, U)
    const float* __restrict__ esrc,  // (B*N,)
    const float* __restrict__ edst,  // (B*N,)
    const int*   __restrict__ edges, // (E, 2) [src, dst]
    float* __restrict__ out)         // (B*N, U)
{
    const int lane = threadIdx.x & 31;
    const int wv   = threadIdx.x >> 5;
    const int gw   = blockIdx.x * 8 + wv;    // node index over B*N
    const int b    = gw >> 14;               // N = 16384
    const int n    = gw & (GAT_N - 1);
    const int eb   = n * GAT_BIN;

    const int srcj = edges[(size_t)(eb + lane) * 2];   // edges[:,0] = src

    float s = esrc[b * GAT_N + srcj] + edst[gw];
    s = (s > 0.f) ? s : 0.2f * s;             // leaky_relu(0.2)
    s = fminf(2.f, fmaxf(-2.f, s));           // clip
    s = expf(s);

    float denom = s;
#pragma unroll
    for (int msk = 16; msk >= 1; msk >>= 1) denom += __shfl_xor(denom, msk);
    const float w = s / denom;

    const float* hb = h + (size_t)b * GAT_N * GAT_U;
    float4 acc = make_float4(0.f, 0.f, 0.f, 0.f);
#pragma unroll 8
    for (int j = 0; j < GAT_BIN; ++j) {
        const float wj = __shfl(w, j);
        const int   sj = __shfl(srcj, j);
        const float4 hv = *(const float4*)(hb + (size_t)sj * GAT_U + lane * 4);
        acc.x += wj * hv.x;
        acc.y += wj * hv.y;
        acc.z += wj * hv.z;
        acc.w += wj * hv.w;
    }
    *(float4*)(out + (size_t)gw * GAT_U + lane * 4) = acc;
}

// ---------------------------------------------------------------------------
extern "C" void kernel_launch(void* const* d_in, const int* in_sizes, int n_in,
                              void* d_out, int out_size, void* d_ws, size_t ws_size,
                              hipStream_t stream) {
    (void)in_sizes; (void)n_in; (void)out_size; (void)ws_size;
    const float* ns    = (const float*)d_in[0];  // node_states (B,N,U) f32
    const int*   edges = (const int*)  d_in[1];  // edges (E,2) i32
    const float* kmat  = (const float*)d_in[2];  // kernel (U,U) f32
    const float* katt  = (const float*)d_in[3];  // kernel_attention (2U,1) f32
    float* out = (float*)d_out;

    float* hbuf = (float*)d_ws;                          // 16 MB
    float* esrc = hbuf + (size_t)GAT_ROWS * GAT_U;       // 128 KB
    float* edst = esrc + GAT_ROWS;                       // 128 KB

    // 512 blocks x 4 waves: each wave = one 16-row strip of h
    gat_gemm_wmma<<<GAT_ROWS / 64, 128, 0, stream>>>(ns, kmat, katt, hbuf, esrc, edst);
    // 4096 blocks x 8 waves: one wave per output node
    gat_aggregate<<<GAT_ROWS / 8, 256, 0, stream>>>(hbuf, esrc, edst, edges, out);
}